// LabelSmoothingLoss_84026740178993
// MI455X (gfx1250) — compile-verified
//
#include <hip/hip_runtime.h>
#include <math.h>

#define BLOCK 256
#define WAVES 8
#define DEPTH 4            // async pipeline depth per wave (power of 2)
#define CHUNK 128          // floats per chunk (32 lanes x float4)
#define EPSF  1e-12f
#define LN2F  0.69314718055994530942f

typedef float v2f __attribute__((ext_vector_type(2)));
typedef float v8f __attribute__((ext_vector_type(8)));

// ---- CDNA5 async global->LDS load (GLOBAL_LOAD_ASYNC_TO_LDS_B128, ASYNCcnt) ----
__device__ __forceinline__ void async_load_b128(unsigned lds_off, const float* gp) {
    asm volatile("global_load_async_to_lds_b128 %0, %1, off"
                 :: "v"(lds_off), "v"((unsigned long long)(uintptr_t)gp)
                 : "memory");
}
#define WAIT_ASYNC(n) asm volatile("s_wait_asynccnt " #n ::: "memory")
#define WAIT_DS0()    asm volatile("s_wait_dscnt 0" ::: "memory")

// ---- wave32 sum-reduction via V_WMMA_F32_16X16X4_F32 (A x ones) ----
// A: lane partial in VGPR0, 0 in VGPR1 -> each A[m][k] slot holds each partial once.
// D[m][n] = rowsum(A[m]); lane L holds D[0..7][L], lane L+16 holds D[8..15][L].
// Sum of 8 D regs + shfl_xor(16) == sum of all 32 partials, in every lane.
__device__ __forceinline__ float wave_reduce_sum(float partial) {
    v2f a; a.x = partial; a.y = 0.0f;
    v2f b; b.x = 1.0f;    b.y = 1.0f;
    v8f c = {};
    c = __builtin_amdgcn_wmma_f32_16x16x4_f32(false, a, false, b,
                                              (short)0, c, false, false);
    float t = c[0] + c[1] + c[2] + c[3] + c[4] + c[5] + c[6] + c[7];
    t += __shfl_xor(t, 16, 32);
    return t;
}

// One block per row: S = sum_v log2(row[v]+eps), then closed-form smoothed KL term.
__global__ __launch_bounds__(BLOCK)
void row_loss_kernel(const float* __restrict__ probs, const int* __restrict__ target,
                     float* __restrict__ row_loss, int V,
                     float sval, float conf, float H1, float H2) {
    __shared__ __align__(16) float stage[WAVES][DEPTH][CHUNK];   // 16 KB
    __shared__ float red[WAVES];

    const int n    = blockIdx.x;
    const int tid  = threadIdx.x;
    const int w    = tid >> 5;
    const int lane = tid & 31;
    const float* __restrict__ row = probs + (size_t)n * (size_t)V;

    // Align body to 16B for b128 async loads.
    int skip = (int)(((16u - ((unsigned)(uintptr_t)row & 15u)) & 15u) >> 2);
    if (skip > V) skip = V;
    const int body      = (V - skip) >> 7;         // number of 128-float chunks
    const int tailStart = skip + (body << 7);

    float partial = 0.0f;

    // Scalar head + tail (direct global loads).
    if (tid < skip) partial += __log2f(row[tid] + EPSF);
    for (int i = tailStart + tid; i < V; i += BLOCK)
        partial += __log2f(row[i] + EPSF);

    // Chunks c = w, w+8, ... assigned to wave w; each wave owns its LDS slots,
    // so no cross-wave synchronization is needed in the streaming loop.
    const int cnt = (body > w) ? ((body - w + (WAVES - 1)) >> 3) : 0;
    const unsigned lds_base = (unsigned)(uintptr_t)&stage[w][0][lane << 2];
    const float* gbase = row + skip + (lane << 2);

    const int pro = (cnt < DEPTH) ? cnt : DEPTH;
    for (int b = 0; b < pro; ++b) {
        const int c = w + (b << 3);
        async_load_b128(lds_base + (unsigned)b * (CHUNK * 4u), gbase + ((size_t)c << 7));
    }

    int j = 0;
    for (; j + DEPTH < cnt; ++j) {                  // steady state: 4 loads in flight
        WAIT_ASYNC(3);                              // oldest chunk landed in LDS
        const int buf = j & (DEPTH - 1);
        const float4 v = *reinterpret_cast<const float4*>(&stage[w][buf][lane << 2]);
        WAIT_DS0();                                 // LDS read done before slot reuse
        const int c2 = w + ((j + DEPTH) << 3);
        async_load_b128(lds_base + (unsigned)buf * (CHUNK * 4u), gbase + ((size_t)c2 << 7));
        partial += __log2f(v.x + EPSF) + __log2f(v.y + EPSF)
                 + __log2f(v.z + EPSF) + __log2f(v.w + EPSF);
    }
    for (; j < cnt; ++j) {                          // drain
        const int pend = cnt - j - 1;
        if      (pend >= 3) { WAIT_ASYNC(3); }
        else if (pend == 2) { WAIT_ASYNC(2); }
        else if (pend == 1) { WAIT_ASYNC(1); }
        else                { WAIT_ASYNC(0); }
        const int buf = j & (DEPTH - 1);
        const float4 v = *reinterpret_cast<const float4*>(&stage[w][buf][lane << 2]);
        partial += __log2f(v.x + EPSF) + __log2f(v.y + EPSF)
                 + __log2f(v.z + EPSF) + __log2f(v.w + EPSF);
    }

    // Block reduction: WMMA wave-sum, then 8 wave totals.
    const float wt = wave_reduce_sum(partial);
    if (lane == 0) red[w] = wt;
    __syncthreads();

    if (tid == 0) {
        const float S2  = red[0] + red[1] + red[2] + red[3]
                        + red[4] + red[5] + red[6] + red[7];
        const float Sln = S2 * LN2F;                 // sum of natural logs over the row
        const int   t   = target[n];
        const int   mi  = V - 100;                   // jnp .at[-100] wraps to V-100
        float kl = 0.0f;
        if (t != -100) {
            const float lpm = __log2f(row[mi] + EPSF) * LN2F;
            if (t == mi) {
                kl = H2 - (sval * (Sln - lpm) + conf * lpm);
            } else {
                const float lpt = __log2f(row[t] + EPSF) * LN2F;
                kl = H1 - (sval * (Sln - lpt - lpm) + conf * lpt);
            }
        }
        row_loss[n] = kl;
    }
}

// Deterministic final reduction of N row losses into d_out[0].
__global__ __launch_bounds__(BLOCK)
void final_reduce_kernel(const float* __restrict__ row_loss, int N,
                         float* __restrict__ out) {
    __shared__ float red[WAVES];
    const int tid = threadIdx.x;
    float p = 0.0f;
    for (int i = tid; i < N; i += BLOCK) p += row_loss[i];
    const float wt = wave_reduce_sum(p);
    if ((tid & 31) == 0) red[tid >> 5] = wt;
    __syncthreads();
    if (tid == 0) {
        out[0] = red[0] + red[1] + red[2] + red[3]
               + red[4] + red[5] + red[6] + red[7];
    }
}

extern "C" void kernel_launch(void* const* d_in, const int* in_sizes, int n_in,
                              void* d_out, int out_size, void* d_ws, size_t ws_size,
                              hipStream_t stream) {
    const float* probs  = (const float*)d_in[0];
    const int*   target = (const int*)d_in[1];
    float*       out    = (float*)d_out;
    float*       ws     = (float*)d_ws;     // N per-row losses

    const int N = in_sizes[1];
    const int V = in_sizes[0] / N;

    // Closed-form entropy constants (double precision on host).
    const double s    = 0.1 / (double)(V - 1);
    const double conf = 1.0 - 0.1;
    const double H1   = (double)(V - 2) * s * log(s) + conf * log(conf); // tgt != V-100
    const double H2   = (double)(V - 1) * s * log(s) + conf * log(conf); // tgt == V-100

    row_loss_kernel<<<N, BLOCK, 0, stream>>>(probs, target, ws, V,
                                             (float)s, (float)conf,
                                             (float)H1, (float)H2);
    final_reduce_kernel<<<1, BLOCK, 0, stream>>>(ws, N, out);
}